// Embedding_52561809768867
// MI455X (gfx1250) — compile-verified
//
#include <hip/hip_runtime.h>

// Embedding gather: out[t, :] = emb[token_ids[t], :]
// B=8, S=4096 -> 32768 tokens; DIM=1024 fp32 (4 KB per row).
// Pure bandwidth problem: ~128 MiB stores + ~98 MiB unique row loads.
// Strategy: 256-thread block handles 8 tokens; each thread moves one float4
// per token (global_load_b128 -> global_store_b128). Loads use default RT
// temporal hint (unique rows ~98MB fit in the 192MB L2, repeats hit);
// stores are non-temporal so the 128MB output stream doesn't evict the
// embedding working set from L2.

typedef __attribute__((ext_vector_type(4))) float v4f;

#ifndef EMB_DIM
#define EMB_DIM 1024
#endif

constexpr int TPB = 256;              // = EMB_DIM / 4 floats per thread
constexpr int TOKENS_PER_BLOCK = 8;   // batch index loads + row loads for MLP

__global__ __launch_bounds__(TPB) void embedding_gather_kernel(
    const int* __restrict__ token_ids,
    const float* __restrict__ emb,
    float* __restrict__ out,
    int n_tokens)
{
    const int t = threadIdx.x;                       // 0..255, one float4 lane of the row
    const int tok_base = blockIdx.x * TOKENS_PER_BLOCK;

    if (tok_base + TOKENS_PER_BLOCK <= n_tokens) {
        // ---- fast path: fully populated block ----
        // 1) all index loads (block-uniform -> scalar loads, one s_wait_kmcnt)
        int toks[TOKENS_PER_BLOCK];
#pragma unroll
        for (int i = 0; i < TOKENS_PER_BLOCK; ++i)
            toks[i] = token_ids[tok_base + i];

        // 2) issue all 8 row loads back-to-back: 8 outstanding global_load_b128
        v4f vals[TOKENS_PER_BLOCK];
#pragma unroll
        for (int i = 0; i < TOKENS_PER_BLOCK; ++i) {
            const v4f* src = (const v4f*)(emb + (size_t)toks[i] * EMB_DIM);
            vals[i] = src[t];
        }

        // 3) stream results out with non-temporal stores (keep L2 for the table)
#pragma unroll
        for (int i = 0; i < TOKENS_PER_BLOCK; ++i) {
            v4f* dst = (v4f*)(out + (size_t)(tok_base + i) * EMB_DIM);
            __builtin_nontemporal_store(vals[i], dst + t);
        }
    } else {
        // ---- tail path (not hit for 32768 tokens, kept for generality) ----
        for (int i = 0; i < TOKENS_PER_BLOCK; ++i) {
            const int tok_idx = tok_base + i;
            if (tok_idx >= n_tokens) break;
            const int tok = token_ids[tok_idx];
            const v4f* src = (const v4f*)(emb + (size_t)tok * EMB_DIM);
            v4f* dst = (v4f*)(out + (size_t)tok_idx * EMB_DIM);
            __builtin_nontemporal_store(src[t], dst + t);
        }
    }
}

extern "C" void kernel_launch(void* const* d_in, const int* in_sizes, int n_in,
                              void* d_out, int out_size, void* d_ws, size_t ws_size,
                              hipStream_t stream) {
    const int*   token_ids = (const int*)d_in[0];     // [8*4096] int32
    const float* emb       = (const float*)d_in[1];   // [50257*1024] fp32
    float*       out       = (float*)d_out;           // [8*4096*1024] fp32

    const int n_tokens = in_sizes[0];                 // 32768
    const int grid = (n_tokens + TOKENS_PER_BLOCK - 1) / TOKENS_PER_BLOCK;

    embedding_gather_kernel<<<grid, TPB, 0, stream>>>(token_ids, emb, out, n_tokens);
}